// PatchDominantGradientOrientation_69466801045728
// MI455X (gfx1250) — compile-verified
//
#include <hip/hip_runtime.h>
#include <math.h>

#define PATCH 32
#define HW (PATCH * PATCH)
#define NBINS 36
#define PPB 16                 // patches per block
#define PHASE_P 8              // patches per TDM phase (double-buffered)
#define PHASE_FLOATS (PHASE_P * HW)   // 8192 floats = 32 KB
#define SMSTRIDE 48            // 36 padded to 3 WMMA N-tiles
#define PI_F 3.14159265358979323846f

typedef unsigned int u32x4 __attribute__((ext_vector_type(4)));
typedef int          i32x4 __attribute__((ext_vector_type(4)));
typedef int          i32x8 __attribute__((ext_vector_type(8)));
typedef float        v2f   __attribute__((ext_vector_type(2)));
typedef float        v8f   __attribute__((ext_vector_type(8)));

#if defined(__gfx1250__) && __has_builtin(__builtin_amdgcn_tensor_load_to_lds) && \
    __has_builtin(__builtin_amdgcn_s_wait_tensorcnt)
#define USE_TDM 1
#else
#define USE_TDM 0
#endif

#if defined(__gfx1250__) && __has_builtin(__builtin_amdgcn_wmma_f32_16x16x4_f32)
#define USE_WMMA 1
#else
#define USE_WMMA 0
#endif

#if USE_TDM
__device__ __forceinline__ void tdm_load_1d(const float* src, uint32_t lds_off, int nfloats)
{
    uint64_t ga = (uint64_t)(uintptr_t)src;
    u32x4 g0;
    g0.x = 1u;                                                 // count=1, flags 0
    g0.y = lds_off;                                            // lds_addr (bytes)
    g0.z = (uint32_t)ga;                                       // global_addr[31:0]
    g0.w = (uint32_t)((ga >> 32) & 0x1FFFFFFu) | 0x80000000u;  // global_addr[56:32] | type=2
    i32x8 g1;
    g1[0] = 0x20000;                           // data_size=2 (4 B)
    g1[1] = (nfloats & 0xFFFF) << 16;          // tensor_dim0 lo16 (bits 63:48)
    g1[2] = ((nfloats >> 16) & 0xFFFF) | (1 << 16); // tensor_dim0 hi16 | tensor_dim1=1
    g1[3] = (nfloats << 16);                   // tile_dim0 (16-bit, 8192 fits)
    g1[4] = 0;                                 // tile_dim1/2 unused
    g1[5] = nfloats;                           // tensor_dim0_stride lo32
    g1[6] = 0;
    g1[7] = 0;
    i32x4 gz = {0, 0, 0, 0};
#if __clang_major__ >= 23
    i32x8 gz8 = {0, 0, 0, 0, 0, 0, 0, 0};
    __builtin_amdgcn_tensor_load_to_lds(g0, g1, gz, gz, gz8, 0);
#else
    __builtin_amdgcn_tensor_load_to_lds(g0, g1, gz, gz, 0);
#endif
}
#endif

// Circulant smoothing matrix: sm[n] = sum_k hist[k] * B[k][n],
// B[k][n] = tap[(k-n+2) mod 36] for the 5-tap discrete Gaussian, else 0.
__device__ __forceinline__ float circ_coef(int k, int n)
{
    if (n >= NBINS) return 0.0f;               // zero-padded N columns 36..47
    int j = k - n + 2;
    if (j < 0) j += NBINS;
    if (j >= NBINS) j -= NBINS;
    float v = 0.0f;
    v = (j == 0 || j == 4) ? 0.119063f : v;    // exp(-t)I_2(t), t=1.6^2, normalized
    v = (j == 1 || j == 3) ? 0.231123f : v;    // exp(-t)I_1(t)
    v = (j == 2)           ? 0.299627f : v;    // exp(-t)I_0(t)
    return v;
}

__device__ __forceinline__ void process_row(const float* __restrict__ p /*1024 floats*/,
                                            float* __restrict__ h /*36 bins*/,
                                            int y, const float* __restrict__ wg, float invS2)
{
    const int ym = (y > 0) ? y - 1 : 0;
    const int yp = (y < PATCH - 1) ? y + 1 : PATCH - 1;
    const float* rm = p + ym * PATCH;
    const float* rc = p + y * PATCH;
    const float* rp = p + yp * PATCH;
    const float wy = wg[y] * invS2;
#pragma unroll 4
    for (int x = 0; x < PATCH; ++x) {
        const int xm = (x > 0) ? x - 1 : 0;
        const int xq = (x < PATCH - 1) ? x + 1 : PATCH - 1;
        float gx = ((rm[xq] - rm[xm]) + 2.0f * (rc[xq] - rc[xm]) + (rp[xq] - rp[xm])) * 0.125f;
        float gy = ((rp[xm] - rm[xm]) + 2.0f * (rp[x] - rm[x]) + (rp[xq] - rm[xq])) * 0.125f;
        float mag = sqrtf(gx * gx + gy * gy + 1e-8f) * (wy * wg[x]);
        float ori = atan2f(gy, gx + 1e-8f) + 2.0f * PI_F;
        float o   = (ori + PI_F) * ((float)NBINS / (2.0f * PI_F));   // in [36, 72]
        float bo0 = floorf(o);
        float wo1 = o - bo0;
        int   b   = (int)bo0 - NBINS;          // [0, 36]
        int   i0  = (b >= NBINS) ? b - NBINS : b;
        int   i1  = i0 + 1;
        i1 = (i1 >= NBINS) ? i1 - NBINS : i1;
        atomicAdd(&h[i0], (1.0f - wo1) * mag); // ds_add_f32
        atomicAdd(&h[i1], wo1 * mag);          // ds_add_f32
    }
}

__global__ __launch_bounds__(256)
void patch_orientation_kernel(const float* __restrict__ patches, float* __restrict__ out)
{
    __shared__ float tile[2 * PHASE_FLOATS];   // 2 x 32 KB double-buffered stage
    __shared__ float hist[PPB * NBINS];        // 16 histograms
    __shared__ float smv[PPB * SMSTRIDE];      // smoothed, padded to 48
    __shared__ float wg[PATCH];
    __shared__ float invS2sh;

    const int tid = threadIdx.x;
    const int blk = blockIdx.x;
    const float* src = patches + (size_t)blk * PPB * HW;

#if USE_TDM
    if (tid < 32) { // wave 0: issue both DMA tiles up front (in-order completion)
        uint32_t lds0 = (uint32_t)(uintptr_t)&tile[0];
        tdm_load_1d(src, lds0, PHASE_FLOATS);
        tdm_load_1d(src + PHASE_FLOATS, lds0 + PHASE_FLOATS * 4, PHASE_FLOATS);
    }
#else
    {
        const float4* s4 = reinterpret_cast<const float4*>(src);
        float4* d4 = reinterpret_cast<float4*>(tile);
        for (int i = tid; i < (2 * PHASE_FLOATS) / 4; i += 256) d4[i] = s4[i];
    }
#endif

    for (int i = tid; i < PPB * NBINS; i += 256) hist[i] = 0.0f;
    if (tid < PATCH) {
        const float sigma = (float)PATCH / 6.0f;
        float x = (float)tid - ((float)PATCH * 0.5f - 0.5f);
        wg[tid] = expf(-x * x / (2.0f * sigma * sigma));
    }
    __syncthreads();
    if (tid == 0) {
        float s = 0.0f;
        for (int i = 0; i < PATCH; ++i) s += wg[i];
        invS2sh = 1.0f / (s * s);
    }
#if USE_TDM
    if (tid < 32) __builtin_amdgcn_s_wait_tensorcnt(1);  // buffer 0 landed
#endif
    __syncthreads();

    // ---- phase 0: patches 0..7 (32 threads/patch, one row each) ----
    {
        const int p = tid >> 5;                  // 0..7
        const int y = tid & 31;
        process_row(&tile[p * HW], &hist[p * NBINS], y, wg, invS2sh);
    }
#if USE_TDM
    if (tid < 32) __builtin_amdgcn_s_wait_tensorcnt(0);  // buffer 1 landed
#endif
    __syncthreads();

    // ---- phase 1: patches 8..15 ----
    {
        const int p = tid >> 5;
        const int y = tid & 31;
        process_row(&tile[PHASE_FLOATS + p * HW], &hist[(PHASE_P + p) * NBINS], y, wg, invS2sh);
    }
    __syncthreads();

    // ---- circular smoothing as SM(16x36) = HIST(16x36) x C(36x36) ----
#if USE_WMMA
    if (tid < 32) {                              // wave 0, EXEC all ones
        const int lane = tid;
        const int m    = lane & 15;              // patch row (M)
        const int half = lane >> 4;              // K sub-pair select
        v8f acc0 = {0, 0, 0, 0, 0, 0, 0, 0};
        v8f acc1 = {0, 0, 0, 0, 0, 0, 0, 0};
        v8f acc2 = {0, 0, 0, 0, 0, 0, 0, 0};
#pragma unroll
        for (int kc = 0; kc < 9; ++kc) {         // K = 36 in chunks of 4
            const int kk = kc * 4 + half * 2;
            v2f A;
            A.x = hist[m * NBINS + kk];
            A.y = hist[m * NBINS + kk + 1];
            v2f B0, B1, B2;
            B0.x = circ_coef(kk,     0 * 16 + m);  B0.y = circ_coef(kk + 1,  0 * 16 + m);
            B1.x = circ_coef(kk,     1 * 16 + m);  B1.y = circ_coef(kk + 1,  1 * 16 + m);
            B2.x = circ_coef(kk,     2 * 16 + m);  B2.y = circ_coef(kk + 1,  2 * 16 + m);
            acc0 = __builtin_amdgcn_wmma_f32_16x16x4_f32(false, A, false, B0, (short)0, acc0, false, false);
            acc1 = __builtin_amdgcn_wmma_f32_16x16x4_f32(false, A, false, B1, (short)0, acc1, false, false);
            acc2 = __builtin_amdgcn_wmma_f32_16x16x4_f32(false, A, false, B2, (short)0, acc2, false, false);
        }
        const float inv = 1.0f / (float)HW;
#pragma unroll
        for (int g = 0; g < 8; ++g) {            // D layout: VGPR g -> row g (+8 for lanes 16..31)
            const int row = g + half * 8;
            smv[row * SMSTRIDE + 0 * 16 + m] = acc0[g] * inv;
            smv[row * SMSTRIDE + 1 * 16 + m] = acc1[g] * inv;
            smv[row * SMSTRIDE + 2 * 16 + m] = acc2[g] * inv;
        }
    }
#else
    for (int i = tid; i < PPB * NBINS; i += 256) {
        const int p = i / NBINS;
        const int n = i - p * NBINS;
        const float* h = &hist[p * NBINS];
        float c  = h[n];
        float l1 = h[(n + NBINS - 1) % NBINS];
        float l2 = h[(n + NBINS - 2) % NBINS];
        float r1 = h[(n + 1) % NBINS];
        float r2 = h[(n + 2) % NBINS];
        smv[p * SMSTRIDE + n] =
            (0.119063f * (l2 + r2) + 0.231123f * (l1 + r1) + 0.299627f * c) * (1.0f / (float)HW);
    }
#endif
    __syncthreads();

    // ---- per-patch argmax + parabolic sub-pixel ----
    if (tid < PPB) {
        const float* s = &smv[tid * SMSTRIDE];
        int idx = 0;
        float best = s[0];
        for (int i = 1; i < NBINS; ++i) {
            float v = s[i];
            if (v > best) { best = v; idx = i; }
        }
        float left  = s[(idx + NBINS - 1) % NBINS];
        float right = s[(idx + 1) % NBINS];
        float cs = 0.5f * (left - right) / (left + right - 2.0f * best);
        out[blk * PPB + tid] = -(2.0f * PI_F * ((float)idx + cs) / (float)NBINS - PI_F);
    }
}

extern "C" void kernel_launch(void* const* d_in, const int* in_sizes, int n_in,
                              void* d_out, int out_size, void* d_ws, size_t ws_size,
                              hipStream_t stream) {
    const float* patches = (const float*)d_in[0];
    float* out = (float*)d_out;
    const int B = in_sizes[0] / HW;          // 32768 patches
    const int nblk = B / PPB;                // 2048 blocks of 16 patches
    patch_orientation_kernel<<<nblk, 256, 0, stream>>>(patches, out);
}